// HouseholderSequence_43018392437484
// MI455X (gfx1250) — compile-verified
//
#include <hip/hip_runtime.h>

typedef __attribute__((ext_vector_type(2))) float v2f;
typedef __attribute__((ext_vector_type(4))) float v4f;
typedef __attribute__((ext_vector_type(8))) float v8f;

#define N_ROWS 262144
#define DDIM   256
#define KREF   32

// Swizzled M layout: k-pairs interleaved per column.
// element (k, n) lives at float index (k>>1)*PAIR_STRIDE + 2*n + (k&1).
// PAIR_STRIDE = 256 cols * 2 + 32 pad floats; 544 mod 64 == 32 so the two
// half-wave B-fragment reads hit disjoint halves of the 64 LDS banks.
#define PAIR_STRIDE   544
#define M_SWZ_FLOATS  (128 * PAIR_STRIDE)   // 69632 floats = 278528 bytes

// ---------------------------------------------------------------------------
// Kernel 1: compose M = H1*H2*...*H32 (each Hk = I - (2/||qk||^2) qk qk^T)
// in LDS, then write it to d_ws in the WMMA-swizzled layout.
// One block, 1024 threads. Thread t owns row r = t&255, column quarter t>>8.
// ---------------------------------------------------------------------------
__global__ void __launch_bounds__(1024)
hh_build_M(const float* __restrict__ q, float* __restrict__ mswz) {
    extern __shared__ float sm[];
    float* Ms    = sm;                 // 256 * 257 floats (row-padded)
    float* qv    = sm + 256 * 257;     // 256
    float* cpart = qv + 256;           // 4 * 256

    const int t  = threadIdx.x;
    const int r  = t & 255;
    const int qq = t >> 8;
    const int j0 = qq * 64;

    // M = I
    for (int j = j0; j < j0 + 64; ++j)
        Ms[r * 257 + j] = (r == j) ? 1.0f : 0.0f;
    __syncthreads();

    for (int k = 0; k < KREF; ++k) {
        if (t < 256) qv[t] = q[k * 256 + t];
        __syncthreads();

        float sq = 0.0f;
        for (int j = 0; j < 256; ++j) sq += qv[j] * qv[j];   // LDS broadcast
        const float s = 2.0f / sq;

        // c = M q  (partial dot over this thread's column quarter)
        float d = 0.0f;
        for (int j = j0; j < j0 + 64; ++j) d += Ms[r * 257 + j] * qv[j];
        cpart[qq * 256 + r] = d;
        __syncthreads();

        const float cr = cpart[r] + cpart[256 + r] + cpart[512 + r] + cpart[768 + r];
        const float cs = cr * s;

        // M -= (s * c) qk^T  (rank-1 update)
        for (int j = j0; j < j0 + 64; ++j)
            Ms[r * 257 + j] -= cs * qv[j];
        __syncthreads();
    }

    // Emit swizzled + bank-padded layout for the GEMM kernel.
    for (int idx = t; idx < 256 * 256; idx += 1024) {
        const int k = idx >> 8, n = idx & 255;
        mswz[(k >> 1) * PAIR_STRIDE + n * 2 + (k & 1)] = Ms[k * 257 + n];
    }
}

// ---------------------------------------------------------------------------
// Kernel 2: out[N,256] = x[N,256] @ M[256,256] with V_WMMA_F32_16X16X4_F32.
// 256 threads = 8 waves; each wave computes 32 rows x 256 cols:
// two A fragments reuse each B fragment -> 32 independent 16x16 f32
// accumulator tiles (256 VGPRs of C), halving LDS B traffic per FLOP.
// M staged once per block into LDS (272 KB) via async global->LDS copies.
// ---------------------------------------------------------------------------
__global__ void __launch_bounds__(256)
hh_gemm(const float* __restrict__ x, const float* __restrict__ mswz,
        float* __restrict__ out, float* __restrict__ logdet) {
    extern __shared__ float lds[];
    const int tid = threadIdx.x;
    const int blk = blockIdx.x;

    // Stage swizzled M (278528 B) into LDS with async b128 copies.
    // Dynamic LDS base is offset 0 here (no static __shared__ in this kernel),
    // so the LDS destination byte address is simply i*16.
    {
        for (int i = tid; i < M_SWZ_FLOATS / 4; i += 256) {
            const uint32_t ldsa = (uint32_t)i * 16u;   // LDS byte address
            const uint32_t goff = (uint32_t)i * 16u;   // byte offset from saddr
            asm volatile("global_load_async_to_lds_b128 %0, %1, %2"
                         :: "v"(ldsa), "v"(goff), "s"(mswz)
                         : "memory");
        }
        asm volatile("s_wait_asynccnt 0x0" ::: "memory");
    }
    // logabsdet = 0 for this block's 256 rows.
    logdet[blk * 256 + tid] = 0.0f;
    __syncthreads();

    const int wave = tid >> 5;      // 0..7
    const int lane = tid & 31;
    const int n    = lane & 15;     // column-within-tile / row-within-band
    const int hi   = lane >> 4;     // half-wave selects K pair (A and B)

    // A fragments: lanes 0-15 read (k,k+1) of rows 0..15 of each band,
    // lanes 16-31 read (k+2,k+3) of the same rows (ISA 16x4 f32 A layout).
    const int rowbase = blk * 256 + wave * 32;
    const float* xrow0 = x + (size_t)(rowbase + n) * DDIM + 2 * hi;
    const float* xrow1 = xrow0 + (size_t)16 * DDIM;

    v8f acc0[16], acc1[16];
#pragma unroll
    for (int c = 0; c < 16; ++c) { acc0[c] = {}; acc1[c] = {}; }

#pragma unroll 2
    for (int k = 0; k < DDIM; k += 4) {
        const v2f a0 = *(const v2f*)(xrow0 + k);
        const v2f a1 = *(const v2f*)(xrow1 + k);
        // B fragments: lanes 0-15 need (M[k][col], M[k+1][col]),
        // lanes 16-31 need (M[k+2][col], M[k+3][col]) -> pair-row (k>>1)+hi.
        const float* bbase = lds + (size_t)((k >> 1) + hi) * PAIR_STRIDE + n * 2;
        v2f b[16];
#pragma unroll
        for (int c = 0; c < 16; ++c)
            b[c] = *(const v2f*)(bbase + c * 32);
#pragma unroll
        for (int c = 0; c < 16; ++c) {
            acc0[c] = __builtin_amdgcn_wmma_f32_16x16x4_f32(
                false, a0, false, b[c], (short)0, acc0[c], false, false);
            acc1[c] = __builtin_amdgcn_wmma_f32_16x16x4_f32(
                false, a1, false, b[c], (short)0, acc1[c], false, false);
        }
    }

    // D tile layout: VGPR v -> row (v + 8*hi), col n (per ISA 16x16 f32 C/D map).
    const int r0 = rowbase + 8 * hi;
#pragma unroll
    for (int c = 0; c < 16; ++c) {
#pragma unroll
        for (int v = 0; v < 8; ++v) {
            out[(size_t)(r0 + v) * DDIM + c * 16 + n]      = acc0[c][v];
            out[(size_t)(r0 + 16 + v) * DDIM + c * 16 + n] = acc1[c][v];
        }
    }
}

// ---------------------------------------------------------------------------
extern "C" void kernel_launch(void* const* d_in, const int* in_sizes, int n_in,
                              void* d_out, int out_size, void* d_ws, size_t ws_size,
                              hipStream_t stream) {
    const float* x = (const float*)d_in[0];   // [N, 256] f32
    const float* q = (const float*)d_in[1];   // [32, 256] f32
    float* out    = (float*)d_out;                       // [N,256]
    float* logdet = out + (size_t)N_ROWS * DDIM;         // [N]
    float* mswz   = (float*)d_ws;                        // 278528 B scratch

    const size_t lds1 = (256 * 257 + 256 + 4 * 256) * sizeof(float); // ~262 KB
    hh_build_M<<<1, 1024, lds1, stream>>>(q, mswz);

    const size_t lds2 = (size_t)M_SWZ_FLOATS * sizeof(float);        // 272 KB
    hh_gemm<<<N_ROWS / 256, 256, lds2, stream>>>(x, mswz, out, logdet);
}